// HashTable_78683800863401
// MI455X (gfx1250) — compile-verified
//
#include <hip/hip_runtime.h>
#include <hip/hip_bf16.h>

// Embedding row gather: out[n, :] = embedding[index[n], :], D = 64 f32.
// Pure data movement -> optimize the memory path, not compute:
//  - 16 lanes per row, one b128 load + one b128 store per lane (256B/row bursts)
//  - RT (default) loads so duplicate rows hit the 192MB L2
//  - NT stores so the 512MB write-once output stream does not evict the table

typedef float v4f __attribute__((ext_vector_type(4)));

#define EMB_D 64  // floats per row
// 16-byte chunks per row = EMB_D*4/16 = 16

__global__ __launch_bounds__(256) void HashTable_gather_kernel(
    const float* __restrict__ emb,   // [U, 64]
    const int*   __restrict__ index, // [N]
    float*       __restrict__ out,   // [N, 64]
    long long n_rows)
{
    long long gid = (long long)blockIdx.x * (long long)blockDim.x + threadIdx.x;
    long long row = gid >> 4;                 // 16 chunks (threads) per row
    int       sub = (int)(gid & 15);          // which 16B chunk within the row
    if (row >= n_rows) return;

    int src = index[row];                     // 16 lanes share one dword: coalesces

    const v4f* __restrict__ srcp =
        (const v4f*)(emb + (long long)src * EMB_D) + sub;
    v4f* __restrict__ dstp =
        (v4f*)(out + row * EMB_D) + sub;

    v4f v = *srcp;                            // regular-temporal: keep table in L2
    __builtin_nontemporal_store(v, dstp);     // NT: don't pollute L2 with output
}

extern "C" void kernel_launch(void* const* d_in, const int* in_sizes, int n_in,
                              void* d_out, int out_size, void* d_ws, size_t ws_size,
                              hipStream_t stream) {
    const float* emb   = (const float*)d_in[0];  // [U*D] f32
    const int*   index = (const int*)d_in[1];    // [N] i32
    float*       out   = (float*)d_out;          // [N*D] f32

    long long n_rows = (long long)in_sizes[1];   // N = number of lookups

    // one thread per 16B chunk: total = n_rows * 16 threads
    long long total_threads = n_rows * 16;
    int block = 256;
    long long grid = (total_threads + block - 1) / block;

    HashTable_gather_kernel<<<(unsigned int)grid, block, 0, stream>>>(
        emb, index, out, n_rows);
}